// YoloNASRAssigner_39152921870791
// MI455X (gfx1250) — compile-verified
//
#include <hip/hip_runtime.h>
#include <stdint.h>

#define BB    8
#define LL    21504
#define NN    128
#define CC    15
#define KTOP  13
#define CHUNK 1024
#define NTHR  256
#define BL    (BB*LL)          // 172032 anchors total
#define WPA   4                // 128 mask bits per anchor -> 4 words
#define LT    (LL/16)          // 1344 anchor tiles (16 anchors each)

typedef float v2f __attribute__((ext_vector_type(2)));
typedef float v8f __attribute__((ext_vector_type(8)));

// ---------------------------------------------------------------- math helpers
__device__ __forceinline__ void covf(float w, float h, float r,
                                     float& a, float& b, float& c) {
  float aa = w * w * (1.0f / 12.0f);
  float bb = h * h * (1.0f / 12.0f);
  float cs = cosf(r), sn = sinf(r);
  a = aa * cs * cs + bb * sn * sn;
  b = aa * sn * sn + bb * cs * cs;
  c = (aa - bb) * cs * sn;
}

// Bhattacharyya-based probabilistic IoU (symmetric in the two boxes).
__device__ __forceinline__ float bhat_iou(float x1, float y1, float a1, float b1, float c1,
                                          float x2, float y2, float a2, float b2, float c2) {
  float A = a1 + a2, Bc = b1 + b2, Cc = c1 + c2;
  float q = A * Bc - Cc * Cc;
  float den = q + 1e-7f;
  float dx = x1 - x2, dy = y1 - y2;
  float t1 = (A * dy * dy + Bc * dx * dx) / den * 0.25f;
  float t2 = Cc * (-dx) * dy / den * 0.5f;
  float det1 = fmaxf(a1 * b1 - c1 * c1, 0.0f);
  float det2 = fmaxf(a2 * b2 - c2 * c2, 0.0f);
  float t3 = 0.5f * logf(q / (4.0f * sqrtf(det1 * det2 + 1e-7f) + 1e-7f) + 1e-7f);
  float bd = fminf(fmaxf(t1 + t2 + t3, 1e-7f), 10.0f);
  return 1.0f - sqrtf(-expm1f(-bd));
}

// ------------------------------------------- CDNA5 async global->LDS (gfx1250)
__device__ __forceinline__ void async_g2l_b128(void* lds, const void* gaddr) {
  uint32_t l = (uint32_t)(uintptr_t)lds;
  asm volatile("global_load_async_to_lds_b128 %0, %1, off"
               :: "v"(l), "v"(gaddr) : "memory");
}
__device__ __forceinline__ void wait_async0() {
  asm volatile("s_wait_asynccnt 0" ::: "memory");
}

// --------------------------------------------------------------------- K0 init
__global__ void k0_init(uint32_t* __restrict__ maskT,
                        uint32_t* __restrict__ max_m,
                        uint32_t* __restrict__ max_i) {
  int i = blockIdx.x * blockDim.x + threadIdx.x;
  if (i < BL * WPA) maskT[i] = 0u;
  if (i < BB * NN) { max_m[i] = 0u; max_i[i] = 0u; }
}

// ---------------------- K_ingts: points-inside-GT-circle mask via WMMA f32 GEMM
// d2(n,l) = |c_n|^2 + |a_l|^2 - 2*c_n.a_l ; cross term is a (16x16, K=4) WMMA
// tile with K padded 2->4. One wave32 per tile. A/B layouts per ISA 7.12.2:
// 32-bit A 16x4: lanes 0-15 hold {K=0,K=1} for M=lane, lanes 16-31 hold K=2,3
// (zero padding here). D: VGPR r -> (M=r, N=lane) / (M=8+r, N=lane-16).
__global__ __launch_bounds__(NTHR) void k_ingts(
    const float* __restrict__ anchor_points,
    const float* __restrict__ gt_rboxes,
    uint16_t* __restrict__ ing16) {
  const int wid  = (blockIdx.x << 3) | (threadIdx.x >> 5);   // wave id = tile id
  const int lane = threadIdx.x & 31;
  const int tpb  = 8 * LT;                                   // tiles per batch
  const int b    = wid / tpb;
  int rem        = wid % tpb;
  const int nt   = rem / LT;
  const int lt   = rem % LT;
  const int n0   = nt * 16, l0 = lt * 16;

  float cx = 0.0f, cy = 0.0f, cn2 = 0.0f, r2v = 0.0f, ax = 0.0f, ay = 0.0f;
  if (lane < 16) {
    const float* g = gt_rboxes + (size_t)(b * NN + n0 + lane) * 5;
    cx = g[0]; cy = g[1];
    float hr = fminf(g[2], g[3]) * 0.5f;
    r2v = hr * hr;
    cn2 = cx * cx + cy * cy;
    ax = anchor_points[2 * (l0 + lane)];
    ay = anchor_points[2 * (l0 + lane) + 1];
  }
  float an2 = __shfl(ax * ax + ay * ay, lane & 15, 32);

  v2f a; a.x = cx; a.y = cy;      // hi lanes carry K=2,3 zero padding
  v2f bv; bv.x = ax; bv.y = ay;
  v8f cz = {};
  v8f d;
#if __has_builtin(__builtin_amdgcn_wmma_f32_16x16x4_f32)
  d = __builtin_amdgcn_wmma_f32_16x16x4_f32(false, a, false, bv,
                                            (short)0, cz, false, false);
#else
  // shuffle fallback (same D layout)
  #pragma unroll
  for (int r = 0; r < 8; ++r) {
    int m = (lane < 16) ? r : (r + 8);
    float cxm = __shfl(cx, m, 32), cym = __shfl(cy, m, 32);
    float axn = __shfl(ax, lane & 15, 32), ayn = __shfl(ay, lane & 15, 32);
    d[r] = cxm * axn + cym * ayn;
  }
#endif

#pragma unroll
  for (int r = 0; r < 8; ++r) {
    int m = (lane < 16) ? r : (r + 8);
    float cn2m = __shfl(cn2, m, 32);
    float r2m  = __shfl(r2v, m, 32);
    float d2   = cn2m + an2 - 2.0f * d[r];
    uint32_t msk = (uint32_t)__ballot(d2 <= r2m);
    if (lane == 0) {
      // ballot bits 0-15 -> GT n0+r, bits 16-31 -> GT n0+8+r (anchors l0..l0+15)
      ing16[(size_t)(b * NN + n0 + r)     * LT + lt] = (uint16_t)(msk & 0xFFFFu);
      ing16[(size_t)(b * NN + n0 + 8 + r) * LT + lt] = (uint16_t)(msk >> 16);
    }
  }
}

// -------------------------------------------------- K1: per-(b,n) row + top-13
__global__ __launch_bounds__(NTHR) void k1_topk(
    const float* __restrict__ pred_scores, const float* __restrict__ pred_rboxes,
    const int* __restrict__ gt_labels, const float* __restrict__ gt_rboxes,
    const float* __restrict__ pad_gt_mask, const uint16_t* __restrict__ ing16,
    uint32_t* __restrict__ maskT) {
  extern __shared__ float smem[];
  float* vals  = smem;                  // LL floats (gated align row)
  float* stage = smem + LL;             // CHUNK*5 floats (pred rboxes chunk)
  __shared__ float red_v[NTHR];
  __shared__ int   red_i[NTHR];
  __shared__ int   topIdx[KTOP];

  const int bn = blockIdx.x;
  const int b  = bn / NN;
  const int n  = bn % NN;
  const int t  = threadIdx.x;

  const float* gtp = gt_rboxes + (size_t)bn * 5;
  const float gx = gtp[0], gy = gtp[1], gw = gtp[2], gh = gtp[3], gr = gtp[4];
  float a1, b1, c1;
  covf(gw, gh, gr, a1, b1, c1);
  const int   lab = gt_labels[bn];
  const float pad = pad_gt_mask[bn];

  const float*    pr  = pred_rboxes + (size_t)b * LL * 5;
  const float*    ps  = pred_scores + (size_t)b * LL * CC;
  const uint16_t* row = ing16 + (size_t)bn * LT;     // in_gts bits for this GT

  for (int c = 0; c < LL / CHUNK; ++c) {
#pragma unroll
    for (int j = 0; j < 5; ++j) {
      int idx = j * (NTHR * 4) + t * 4;                      // 16B aligned
      async_g2l_b128(stage + idx, pr + (size_t)c * CHUNK * 5 + idx);
    }
    wait_async0();
    __syncthreads();

    for (int i = t; i < CHUNK; i += NTHR) {
      int l = c * CHUNK + i;
      float px = stage[i * 5 + 0], py = stage[i * 5 + 1];
      float pw = stage[i * 5 + 2], ph = stage[i * 5 + 3], pa = stage[i * 5 + 4];
      float a2, b2, c2;
      covf(pw, ph, pa, a2, b2, c2);
      float iou = bhat_iou(gx, gy, a1, b1, c1, px, py, a2, b2, c2);
      float cls = ps[(size_t)l * CC + lab];
      float i2 = iou * iou;
      float al = cls * i2 * i2 * i2;                 // cls^1 * iou^6
      float ing = (float)((row[l >> 4] >> (l & 15)) & 1u);
      vals[l] = al * ing;
    }
    __syncthreads();
  }

  // 13 rounds of block-wide argmax; ties broken toward the lowest index.
  for (int r = 0; r < KTOP; ++r) {
    float bv = -1.0f;
    int bi = 0;
    for (int i = t; i < LL; i += NTHR) {
      float v = vals[i];
      if (v > bv) { bv = v; bi = i; }               // strict > keeps lowest idx
    }
    red_v[t] = bv; red_i[t] = bi;
    __syncthreads();
    for (int s = NTHR / 2; s > 0; s >>= 1) {
      if (t < s) {
        float ov = red_v[t + s]; int oi = red_i[t + s];
        if (ov > red_v[t] || (ov == red_v[t] && oi < red_i[t])) {
          red_v[t] = ov; red_i[t] = oi;
        }
      }
      __syncthreads();
    }
    if (t == 0) { topIdx[r] = red_i[0]; vals[red_i[0]] = -2.0f; }
    __syncthreads();
  }

  // mask_positive bit = in_topk & in_gts & pad (transposed: 128 bits/anchor).
  if (t < KTOP && pad > 0.0f) {
    int wi = topIdx[t];
    if ((row[wi >> 4] >> (wi & 15)) & 1u) {
      atomicOr(&maskT[((size_t)b * LL + wi) * WPA + (n >> 5)], 1u << (n & 31));
    }
  }
}

// ---------------------------- K2: per-anchor multi resolution + row max accum
__global__ __launch_bounds__(NTHR) void k2_resolve(
    const float* __restrict__ pred_scores, const float* __restrict__ pred_rboxes,
    const float* __restrict__ gt_rboxes, const int* __restrict__ gt_labels,
    const uint32_t* __restrict__ maskT,
    int* __restrict__ code, float* __restrict__ aval_arr,
    uint32_t* __restrict__ max_m, uint32_t* __restrict__ max_i) {
  __shared__ float sgt[NN * 5];
  __shared__ int   slab[NN];
  const int t   = threadIdx.x;
  const int gid = blockIdx.x * NTHR + t;        // == b*LL + l
  const int b   = blockIdx.x / (LL / NTHR);     // LL % NTHR == 0 -> uniform b

  if (t * 4 < NN * 5)
    async_g2l_b128(sgt + t * 4, gt_rboxes + (size_t)b * NN * 5 + t * 4);
  if (t < NN) slab[t] = gt_labels[b * NN + t];
  wait_async0();
  __syncthreads();

  const float* pp = pred_rboxes + (size_t)gid * 5;
  float px = pp[0], py = pp[1], pw = pp[2], ph = pp[3], pa = pp[4];
  float a2, b2, c2;
  covf(pw, ph, pa, a2, b2, c2);

  uint32_t w[WPA];
#pragma unroll
  for (int i = 0; i < WPA; ++i) w[i] = maskT[(size_t)gid * WPA + i];
  int mps0 = __popc(w[0]) + __popc(w[1]) + __popc(w[2]) + __popc(w[3]);

  int agi = 0;
  bool pos = false;
  if (mps0 > 1) {
    float best = -1.0f;
    int bi = 0;
    for (int n = 0; n < NN; ++n) {
      const float* g = sgt + n * 5;
      float ga, gb, gc;
      covf(g[2], g[3], g[4], ga, gb, gc);
      float iou = bhat_iou(g[0], g[1], ga, gb, gc, px, py, a2, b2, c2);
      if (iou > best) { best = iou; bi = n; }
    }
    agi = bi; pos = true;
  } else if (mps0 == 1) {
#pragma unroll
    for (int i = 0; i < WPA; ++i)
      if (w[i]) { agi = i * 32 + (__ffs(w[i]) - 1); break; }
    pos = true;
  }

  float av = 0.0f;
  if (pos) {
    const float* g = sgt + agi * 5;
    float ga, gb, gc;
    covf(g[2], g[3], g[4], ga, gb, gc);
    float iou = bhat_iou(g[0], g[1], ga, gb, gc, px, py, a2, b2, c2);
    float cls = pred_scores[(size_t)gid * CC + slab[agi]];
    float i2 = iou * iou;
    av = cls * i2 * i2 * i2;
    atomicMax(&max_m[b * NN + agi], __float_as_uint(av));
    atomicMax(&max_i[b * NN + agi], __float_as_uint(iou));
  }
  code[gid]     = pos ? (agi + 1) : 0;
  aval_arr[gid] = av;
}

// ------------------------------------------------------- K3: emit all outputs
__global__ __launch_bounds__(NTHR) void k3_write(
    const float* __restrict__ gt_rboxes, const int* __restrict__ gt_labels,
    const int* __restrict__ gt_crowd,
    const int* __restrict__ code, const float* __restrict__ aval_arr,
    const uint32_t* __restrict__ max_m, const uint32_t* __restrict__ max_i,
    const int* __restrict__ bg_ptr, float* __restrict__ out) {
  int gid = blockIdx.x * blockDim.x + threadIdx.x;
  if (gid >= BL) return;
  int b  = gid / LL;
  int cd = code[gid];
  bool pos = cd > 0;
  int agi  = pos ? (cd - 1) : 0;
  int gi   = b * NN + agi;

  float* out_lab = out;
  float* out_rb  = out + (size_t)BL;
  float* out_sc  = out + (size_t)BL * 6;
  float* out_gi  = out + (size_t)BL * (6 + CC);
  float* out_cr  = out + (size_t)BL * (7 + CC);

  int bg  = *bg_ptr;
  int lab = gt_labels[gi];
  out_lab[gid] = (float)(pos ? lab : bg);
#pragma unroll
  for (int j = 0; j < 5; ++j)
    out_rb[(size_t)gid * 5 + j] = gt_rboxes[(size_t)gi * 5 + j];

  float s = 0.0f;
  if (pos) {
    float mm = __uint_as_float(max_m[gi]);
    float mi = __uint_as_float(max_i[gi]);
    s = aval_arr[gid] / (mm + 1e-9f) * mi;
  }
#pragma unroll
  for (int c = 0; c < CC; ++c)
    out_sc[(size_t)gid * CC + c] = (pos && c == lab) ? s : 0.0f;
  out_gi[gid] = (float)gi;
  out_cr[gid] = (gt_crowd[gi] != 0) ? 1.0f : 0.0f;
}

// ----------------------------------------------------------------- launcher
extern "C" void kernel_launch(void* const* d_in, const int* in_sizes, int n_in,
                              void* d_out, int out_size, void* d_ws, size_t ws_size,
                              hipStream_t stream) {
  const float* pred_scores   = (const float*)d_in[0];
  const float* pred_rboxes   = (const float*)d_in[1];
  const float* anchor_points = (const float*)d_in[2];
  const int*   gt_labels     = (const int*)d_in[3];
  const float* gt_rboxes     = (const float*)d_in[4];
  const int*   gt_crowd      = (const int*)d_in[5];
  const float* pad_gt_mask   = (const float*)d_in[6];
  const int*   bg_ptr        = (const int*)d_in[7];

  // Workspace layout (~7 MB)
  uint8_t* p = (uint8_t*)d_ws;
  uint32_t* maskT = (uint32_t*)p;  p += (size_t)BL * WPA * 4;
  uint16_t* ing16 = (uint16_t*)p;  p += (size_t)BB * NN * LT * 2;
  int*      codeA = (int*)p;       p += (size_t)BL * 4;
  float*    avalA = (float*)p;     p += (size_t)BL * 4;
  uint32_t* max_m = (uint32_t*)p;  p += (size_t)BB * NN * 4;
  uint32_t* max_i = (uint32_t*)p;

  k0_init<<<(BL * WPA + NTHR - 1) / NTHR, NTHR, 0, stream>>>(maskT, max_m, max_i);

  // in_gts bits via WMMA pairwise-distance GEMM: 86016 tiles, 8 waves/block
  const int ntiles = BB * 8 * LT;
  k_ingts<<<ntiles / 8, NTHR, 0, stream>>>(anchor_points, gt_rboxes, ing16);

  const size_t dynLDS = (size_t)(LL + CHUNK * 5) * sizeof(float);  // 104 KB < 320 KB/WGP
  k1_topk<<<BB * NN, NTHR, dynLDS, stream>>>(pred_scores, pred_rboxes, gt_labels,
                                             gt_rboxes, pad_gt_mask, ing16, maskT);

  k2_resolve<<<BL / NTHR, NTHR, 0, stream>>>(pred_scores, pred_rboxes, gt_rboxes,
                                             gt_labels, maskT, codeA, avalA, max_m, max_i);

  k3_write<<<(BL + NTHR - 1) / NTHR, NTHR, 0, stream>>>(gt_rboxes, gt_labels, gt_crowd,
                                                        codeA, avalA, max_m, max_i,
                                                        bg_ptr, (float*)d_out);
}